// SelfAttention_12189117186170
// MI455X (gfx1250) — compile-verified
//
#include <hip/hip_runtime.h>
#include <math.h>
#include <stdint.h>

typedef __attribute__((ext_vector_type(2))) float v2f;
typedef __attribute__((ext_vector_type(8))) float v8f;
typedef __attribute__((ext_vector_type(4))) int   v4i;

#ifndef __has_builtin
#define __has_builtin(x) 0
#endif

#define AS1 __attribute__((address_space(1)))
#define AS3 __attribute__((address_space(3)))

#define B      4
#define DIM    4096
#define HQ     32
#define KVH    8
#define HD     128
#define BLOCK  256
#define N_REP  4
#define QKV_COLS 6144      /* 4096 q + 1024 k + 1024 v */
#define CHUNK  512
#define NCHUNK 8
#define REC    520         /* partial record: 4 m + 4 l + 4*128 acc */

/* workspace offsets in floats */
#define WS_QBUF  0
#define WS_KBUF  (WS_QBUF + B*HQ*HD)            /* 16384 */
#define WS_VBUF  (WS_KBUF + B*KVH*HD)
#define WS_QKVP  (WS_VBUF + B*KVH*HD)
#define WS_ATTN  (WS_QKVP + NCHUNK*B*QKV_COLS)
#define WS_ATTNP (WS_ATTN + B*HQ*HD)
#define WS_OPP   (WS_ATTNP + 32*64*REC)

/* ---------------- kernel 1: QKV GEMV partials (split over contraction) ------- */
__global__ void qkv_partial(const float* __restrict__ x,
                            const float* __restrict__ wq,
                            const float* __restrict__ wk,
                            const float* __restrict__ wv,
                            float* __restrict__ part) {
  __shared__ float x_lds[B][CHUNK];
  const int t = threadIdx.x;
  const int jblk = blockIdx.x, chunk = blockIdx.y;
  for (int i = t; i < B*CHUNK; i += 256)
    x_lds[i >> 9][i & 511] = x[(size_t)(i >> 9)*DIM + chunk*CHUNK + (i & 511)];
  __syncthreads();
  const int j = jblk*256 + t;
  const float* W; int col, stride;
  if (j < 4096)      { W = wq; col = j;        stride = 4096; }
  else if (j < 5120) { W = wk; col = j - 4096; stride = 1024; }
  else               { W = wv; col = j - 5120; stride = 1024; }
  const float* wp = W + (size_t)chunk*CHUNK*stride + col;
  float a0 = 0.f, a1 = 0.f, a2 = 0.f, a3 = 0.f;
  for (int ii = 0; ii < CHUNK; ++ii) {
    __builtin_prefetch(wp + (size_t)(ii + 8)*stride, 0, 1);
    float w = wp[(size_t)ii*stride];
    a0 = fmaf(x_lds[0][ii], w, a0);
    a1 = fmaf(x_lds[1][ii], w, a1);
    a2 = fmaf(x_lds[2][ii], w, a2);
    a3 = fmaf(x_lds[3][ii], w, a3);
  }
  part[(size_t)(chunk*B + 0)*QKV_COLS + j] = a0;
  part[(size_t)(chunk*B + 1)*QKV_COLS + j] = a1;
  part[(size_t)(chunk*B + 2)*QKV_COLS + j] = a2;
  part[(size_t)(chunk*B + 3)*QKV_COLS + j] = a3;
}

/* ---------------- kernel 2: reduce partials + RoPE ------------------------- */
__global__ void qkv_rope(const float* __restrict__ part,
                         const float* __restrict__ fcos,
                         const float* __restrict__ fsin,
                         float* __restrict__ qbuf,
                         float* __restrict__ kbuf,
                         float* __restrict__ vbuf) {
  const int idx = blockIdx.x*256 + threadIdx.x;    /* 0 .. 24575 */
  if (idx >= B*QKV_COLS) return;
  const int b = idx / QKV_COLS, j = idx % QKV_COLS;
  const int jp = j ^ 1;
  float self = 0.f, partner = 0.f;
  for (int c = 0; c < NCHUNK; ++c) {
    self    += part[(size_t)(c*B + b)*QKV_COLS + j];
    partner += part[(size_t)(c*B + b)*QKV_COLS + jp];
  }
  float val;
  if (j < 5120) {                                   /* q or k: apply RoPE */
    const int i = (j & 127) >> 1;
    const float cc = fcos[i], ss = fsin[i];
    if ((j & 1) == 0) val = self*cc - partner*ss;   /* even: xr */
    else              val = partner*ss + self*cc;   /* odd : xi */
  } else val = self;
  if (j < 4096)       qbuf[(size_t)b*4096 + j]          = val;
  else if (j < 5120)  kbuf[(size_t)b*1024 + (j - 4096)] = val;
  else                vbuf[(size_t)b*1024 + (j - 5120)] = val;
}

/* ---------------- kernel 3: flash-style attention partials (WMMA scores) ---- */
/* grid.x = 64 (sel-block s = x>>1, half = x&1 -> 128 keys each)
   grid.y = 32 (b = y>>3, kv-group g = y&7); block = 128 threads (4 waves)     */
__global__ void attn_partial(const float* __restrict__ qbuf,
                             const float* __restrict__ kbuf,
                             const float* __restrict__ vbuf,
                             const float* __restrict__ k_cache,
                             const float* __restrict__ v_cache,
                             const int*   __restrict__ blk_idx,
                             const int*   __restrict__ sp_ptr,
                             float* __restrict__ part) {
  __shared__ float q_lds[16*132];
  __shared__ float kv_lds[64*132];
  __shared__ float p_lds[4*128];
  __shared__ float red[8];

  const int t = threadIdx.x;
  const int lane = t & 31, wave = t >> 5;
  const int s = blockIdx.x >> 1, half = blockIdx.x & 1;
  const int b = blockIdx.y >> 3, g = blockIdx.y & 7;
  const int blk = blk_idx[s];
  const int sp = sp_ptr[0];
  const int blk_upd = sp / BLOCK, pos_upd = sp % BLOCK;

  /* stage 4 real q rows (heads g*4 .. g*4+3), zero-pad rows 4..15 */
  for (int i = t; i < 16*128; i += 128) {
    int r = i >> 7, d = i & 127;
    float v = 0.f;
    if (r < N_REP) v = qbuf[(size_t)(b*HQ + g*N_REP + r)*HD + d];
    q_lds[r*132 + d] = v;
  }

  float pv0 = 0.f, pv1 = 0.f, pv2 = 0.f, pv3 = 0.f;

  for (int pass = 0; pass < 2; ++pass) {            /* 0: Q.K^T  1: P.V */
    const float* cache = pass ? v_cache : k_cache;
    const float* fresh = pass ? vbuf    : kbuf;
    for (int chunk = 0; chunk < 2; ++chunk) {
      const int key0 = half*128 + chunk*64;
      /* stage 64 rows x 128 floats into padded LDS (b128 granularity) */
#if __has_builtin(__builtin_amdgcn_global_load_async_to_lds_b128)
      for (int i = 0; i < 16; ++i) {
        int f = i*128 + t;
        int kl = f >> 5, w = f & 31;
        size_t idx = ((((size_t)blk*BLOCK + key0 + kl)*B + b)*KVH + g)*(size_t)HD + w*4;
        AS1 v4i* gp = (AS1 v4i*)(uintptr_t)(cache + idx);
        AS3 v4i* lp = (AS3 v4i*)(uint32_t)(uintptr_t)&kv_lds[kl*132 + w*4];
        __builtin_amdgcn_global_load_async_to_lds_b128(gp, lp, 0, 0);
      }
#if __has_builtin(__builtin_amdgcn_s_wait_asynccnt)
      __builtin_amdgcn_s_wait_asynccnt(0);
#else
      asm volatile("s_wait_asynccnt 0x0" ::: "memory");
#endif
#else
      for (int i = 0; i < 16; ++i) {
        int f = i*128 + t;
        int kl = f >> 5, w = f & 31;
        size_t idx = ((((size_t)blk*BLOCK + key0 + kl)*B + b)*KVH + g)*(size_t)HD + w*4;
        *(float4*)&kv_lds[kl*132 + w*4] = *(const float4*)(cache + idx);
      }
#endif
      __syncthreads();
      /* functional cache update: overwrite the freshly-written k/v row */
      if (blk == blk_upd && pos_upd >= key0 && pos_upd < key0 + 64)
        kv_lds[(pos_upd - key0)*132 + t] = fresh[(size_t)(b*KVH + g)*HD + t];
      __syncthreads();

      if (pass == 0) {
        /* each wave: one 16-key tile; D = Q(16x128) . K^T via 32 k-chunks,
           two independent accumulation chains to break the D->C RAW chain  */
        v8f c0 = {}, c1 = {};
        const int m  = lane & 15;
        const int off = (lane < 16) ? 0 : 2;        /* A/B f32 16x4 layout */
        const int n0 = wave*16;
        const float* qrow = &q_lds[m*132];
        const float* krow = &kv_lds[(n0 + m)*132];
        for (int kk = 0; kk < 32; kk += 2) {
          v2f a0v, b0v, a1v, b1v;
          a0v.x = qrow[4*kk + off];      a0v.y = qrow[4*kk + off + 1];
          b0v.x = krow[4*kk + off];      b0v.y = krow[4*kk + off + 1];
          a1v.x = qrow[4*kk + 4 + off];  a1v.y = qrow[4*kk + 4 + off + 1];
          b1v.x = krow[4*kk + 4 + off];  b1v.y = krow[4*kk + 4 + off + 1];
          c0 = __builtin_amdgcn_wmma_f32_16x16x4_f32(false, a0v, false, b0v,
                                                     (short)0, c0, false, false);
          c1 = __builtin_amdgcn_wmma_f32_16x16x4_f32(false, a1v, false, b1v,
                                                     (short)0, c1, false, false);
        }
        v8f c = c0 + c1;
        if (lane < 16) {                            /* rows M=0..3 live here */
          const float sc = 0.088388347648318447f;   /* 1/sqrt(128) */
          const int col = chunk*64 + n0 + m;
          p_lds[0*128 + col] = c[0]*sc;
          p_lds[1*128 + col] = c[1]*sc;
          p_lds[2*128 + col] = c[2]*sc;
          p_lds[3*128 + col] = c[3]*sc;
        }
      } else {
        /* P.V : thread t owns dim d = t for all 4 rep rows */
        for (int n = 0; n < 64; ++n) {
          float v = kv_lds[n*132 + t];
          int col = chunk*64 + n;
          pv0 = fmaf(p_lds[0*128 + col], v, pv0);
          pv1 = fmaf(p_lds[1*128 + col], v, pv1);
          pv2 = fmaf(p_lds[2*128 + col], v, pv2);
          pv3 = fmaf(p_lds[3*128 + col], v, pv3);
        }
      }
      __syncthreads();
    }
    if (pass == 0) {
      /* local softmax stats over this WG's 128 keys */
      if (t < 4) {
        float mx = -3.4e38f;
        for (int n = 0; n < 128; ++n) mx = fmaxf(mx, p_lds[t*128 + n]);
        float l = 0.f;
        for (int n = 0; n < 128; ++n) {
          float e = expf(p_lds[t*128 + n] - mx);
          p_lds[t*128 + n] = e;
          l += e;
        }
        red[t] = mx; red[4 + t] = l;
      }
      __syncthreads();
    }
  }

  const size_t rec = ((size_t)blockIdx.y*64 + blockIdx.x)*REC;
  if (t < 8) part[rec + t] = red[t];
  part[rec + 8 + 0*128 + t] = pv0;
  part[rec + 8 + 1*128 + t] = pv1;
  part[rec + 8 + 2*128 + t] = pv2;
  part[rec + 8 + 3*128 + t] = pv3;
}

/* ---------------- kernel 4: combine flash partials -------------------------- */
__global__ void attn_reduce(const float* __restrict__ part,
                            float* __restrict__ attn) {
  const int bgr = blockIdx.x;          /* ((b*8+g)*4 + r) */
  const int r = bgr & 3, bg = bgr >> 2;
  const int d = threadIdx.x;
  float M = -3.4e38f;
  for (int s = 0; s < 64; ++s)
    M = fmaxf(M, part[((size_t)bg*64 + s)*REC + r]);
  float L = 0.f, acc = 0.f;
  for (int s = 0; s < 64; ++s) {
    const float* rp = part + ((size_t)bg*64 + s)*REC;
    float w = expf(rp[r] - M);
    L   = fmaf(rp[4 + r], w, L);
    acc = fmaf(rp[8 + r*128 + d], w, acc);
  }
  const int b = bg >> 3, g = bg & 7;
  attn[(size_t)(b*HQ + g*N_REP + r)*HD + d] = acc / L;
}

/* ---------------- kernel 5/6: output projection ----------------------------- */
__global__ void oproj_partial(const float* __restrict__ attn,
                              const float* __restrict__ wo,
                              float* __restrict__ part) {
  __shared__ float a_lds[B][CHUNK];
  const int t = threadIdx.x;
  const int jblk = blockIdx.x, chunk = blockIdx.y;
  for (int i = t; i < B*CHUNK; i += 256)
    a_lds[i >> 9][i & 511] = attn[(size_t)(i >> 9)*DIM + chunk*CHUNK + (i & 511)];
  __syncthreads();
  const int j = jblk*256 + t;
  const float* wp = wo + (size_t)chunk*CHUNK*DIM + j;
  float a0 = 0.f, a1 = 0.f, a2 = 0.f, a3 = 0.f;
  for (int ii = 0; ii < CHUNK; ++ii) {
    __builtin_prefetch(wp + (size_t)(ii + 8)*DIM, 0, 1);
    float w = wp[(size_t)ii*DIM];
    a0 = fmaf(a_lds[0][ii], w, a0);
    a1 = fmaf(a_lds[1][ii], w, a1);
    a2 = fmaf(a_lds[2][ii], w, a2);
    a3 = fmaf(a_lds[3][ii], w, a3);
  }
  part[(size_t)(chunk*B + 0)*DIM + j] = a0;
  part[(size_t)(chunk*B + 1)*DIM + j] = a1;
  part[(size_t)(chunk*B + 2)*DIM + j] = a2;
  part[(size_t)(chunk*B + 3)*DIM + j] = a3;
}

__global__ void oproj_reduce(const float* __restrict__ part,
                             float* __restrict__ out) {
  const int idx = blockIdx.x*256 + threadIdx.x;     /* 0 .. 16383 */
  const int b = idx / DIM, j = idx % DIM;
  float acc = 0.f;
  for (int c = 0; c < NCHUNK; ++c)
    acc += part[(size_t)(c*B + b)*DIM + j];
  out[(size_t)b*DIM + j] = acc;
}

/* ---------------- host-side launcher --------------------------------------- */
extern "C" void kernel_launch(void* const* d_in, const int* in_sizes, int n_in,
                              void* d_out, int out_size, void* d_ws, size_t ws_size,
                              hipStream_t stream) {
  const float* x  = (const float*)d_in[0];
  const float* wq = (const float*)d_in[1];
  const float* wk = (const float*)d_in[2];
  const float* wv = (const float*)d_in[3];
  const float* wo = (const float*)d_in[4];
  const float* kc = (const float*)d_in[5];
  const float* vc = (const float*)d_in[6];
  const float* fc = (const float*)d_in[7];
  const float* fs = (const float*)d_in[8];
  const int*   bi = (const int*)d_in[9];
  const int*   sp = (const int*)d_in[10];
  float* ws = (float*)d_ws;
  float* qbuf  = ws + WS_QBUF;
  float* kbuf  = ws + WS_KBUF;
  float* vbuf  = ws + WS_VBUF;
  float* qkvp  = ws + WS_QKVP;
  float* attn  = ws + WS_ATTN;
  float* attnp = ws + WS_ATTNP;
  float* opp   = ws + WS_OPP;

  qkv_partial <<<dim3(QKV_COLS/256, NCHUNK), 256, 0, stream>>>(x, wq, wk, wv, qkvp);
  qkv_rope    <<<dim3((B*QKV_COLS + 255)/256), 256, 0, stream>>>(qkvp, fc, fs, qbuf, kbuf, vbuf);
  attn_partial<<<dim3(64, 32), 128, 0, stream>>>(qbuf, kbuf, vbuf, kc, vc, bi, sp, attnp);
  attn_reduce <<<dim3(B*KVH*N_REP), HD, 0, stream>>>(attnp, attn);
  oproj_partial<<<dim3(DIM/256, NCHUNK), 256, 0, stream>>>(attn, wo, opp);
  oproj_reduce <<<dim3(B*DIM/256), 256, 0, stream>>>(opp, (float*)d_out);
}